// DGCNNpp_19387482374365
// MI455X (gfx1250) — compile-verified
//
#include <hip/hip_runtime.h>
#include <math.h>

#define KNN   20
#define NEGL  -1.0e9f
#define INVBN 0.9999950000374998f   // 1/sqrt(1+1e-5)
#define PTS   4
#define COLS  (PTS * KNN)           // 80 = 5 WMMA column tiles of 16

typedef float v2f __attribute__((ext_vector_type(2)));
typedef float v8f __attribute__((ext_vector_type(8)));

__device__ __forceinline__ v8f wmma4(v2f a, v2f b, v8f c) {
  // V_WMMA_F32_16X16X4_F32 : D = A(16x4,f32) * B(4x16,f32) + C(16x16,f32)
  return __builtin_amdgcn_wmma_f32_16x16x4_f32(false, a, false, b, (short)0, c, false, false);
}

__device__ __forceinline__ float lrelu(float x) { return x > 0.f ? x : 0.2f * x; }

__device__ __forceinline__ unsigned pcg_hash(unsigned v) {
  v = v * 747796405u + 2891336453u;
  unsigned w = ((v >> ((v >> 28u) + 4u)) ^ v) * 277803737u;
  return (w >> 22u) ^ w;
}

__device__ __forceinline__ float gumbel_noise(unsigned h) {
  float u = ((float)(h >> 8) + 0.5f) * (1.0f / 16777216.0f);  // (0,1)
  float e = -logf(u);                                         // Exp(1)
  return -logf(e + 1e-30f);                                   // gumbel
}

// ---------------------------------------------------------------------------
// kNN: one workgroup per (b,n) query row. Distances in LDS (b128 traffic),
// 20 argmax passes. dist[j] = 2*dot(q,x_j) - |x_j|^2 (ranking-invariant shift)
// ---------------------------------------------------------------------------
template <int C>
__global__ void __launch_bounds__(256)
knn_kernel(const float* __restrict__ feat, long batchStride, int N,
           int* __restrict__ idxOut) {
  __shared__ float q[C];
  __shared__ float dist[4096];
  __shared__ float rv[256];
  __shared__ int   ri[256];
  const int b = blockIdx.x / N;
  const int n = blockIdx.x % N;
  const int tid = threadIdx.x;
  const float* f = feat + (long)b * batchStride;

  if (tid < C) q[tid] = f[(long)tid * N + n];
  __syncthreads();

  for (int j = tid * 4; j < N; j += 1024) {   // float4 per thread
    float dx = 0.f, dy = 0.f, dz = 0.f, dw = 0.f;
    float fx = 0.f, fy = 0.f, fz = 0.f, fw_ = 0.f;
#pragma unroll 4
    for (int c = 0; c < C; ++c) {
      float qc = q[c];
      float4 v = *(const float4*)&f[(long)c * N + j];
      dx += qc * v.x; dy += qc * v.y; dz += qc * v.z; dw += qc * v.w;
      fx += v.x * v.x; fy += v.y * v.y; fz += v.z * v.z; fw_ += v.w * v.w;
    }
    float4 d;
    d.x = 2.f * dx - fx; d.y = 2.f * dy - fy;
    d.z = 2.f * dz - fz; d.w = 2.f * dw - fw_;
    *(float4*)&dist[j] = d;
  }
  __syncthreads();

  for (int kk = 0; kk < KNN; ++kk) {
    float bv = -3.0e38f; int bi = N;
    for (int j = tid * 4; j < N; j += 1024) {  // ascending scan -> lowest-index ties
      float4 v = *(const float4*)&dist[j];
      if (v.x > bv) { bv = v.x; bi = j; }
      if (v.y > bv) { bv = v.y; bi = j + 1; }
      if (v.z > bv) { bv = v.z; bi = j + 2; }
      if (v.w > bv) { bv = v.w; bi = j + 3; }
    }
    rv[tid] = bv; ri[tid] = bi;
    __syncthreads();
    for (int s = 128; s > 0; s >>= 1) {
      if (tid < s) {
        float ov = rv[tid + s]; int oi = ri[tid + s];
        if (ov > rv[tid] || (ov == rv[tid] && oi < ri[tid])) { rv[tid] = ov; ri[tid] = oi; }
      }
      __syncthreads();
    }
    if (tid == 0) {
      idxOut[((long)b * N + n) * KNN + kk] = ri[0];
      dist[ri[0]] = -3.0e38f;
    }
    __syncthreads();
  }
}

// ---------------------------------------------------------------------------
// Fused EdgeConv block (compile-time C / two_trans):
//   gather edges -> LDS, t-branch convs via f32 WMMA (A reused across 2-3
//   column tiles per wave), gumbel gate + softmax attention via VALU, weighted
//   k-reduction via ds_add_f32. One WG = PTS=4 points = 80 GEMM columns.
// Wave w (scalarized): M-tile = w&3 ; column tiles c0 = (w>>2)*16 + {0,32[,64]}
// ---------------------------------------------------------------------------
template <int C, bool TWO_TRANS>
__global__ void __launch_bounds__(256)
edge_block_kernel(const float* __restrict__ fin, long finBS,
                  const int* __restrict__ idx,
                  const float* __restrict__ tw1, const float* __restrict__ tg1, const float* __restrict__ tb1,
                  const float* __restrict__ tw2, const float* __restrict__ tg2, const float* __restrict__ tb2,
                  const float* __restrict__ gw1, const float* __restrict__ gg1, const float* __restrict__ gb1,
                  const float* __restrict__ gw2, const float* __restrict__ gbias,
                  const float* __restrict__ aw1, const float* __restrict__ ag1, const float* __restrict__ ab1,
                  const float* __restrict__ aw2, const float* __restrict__ abias,
                  float* __restrict__ out, long outBS, int N, unsigned seed) {
  constexpr int TWOC = 2 * C;
  constexpr int KP   = (TWOC + 3) & ~3;          // K padded to multiple of 4
  constexpr bool PADDED = (KP != TWOC);

  __shared__ float E[KP * COLS];                  // edge tile [KP x 80]
  __shared__ float T[TWO_TRANS ? 64 * COLS : 1];  // conv1 output (if 2 convs)
  __shared__ float W1p[PADDED ? 64 * KP : 1];     // zero-padded tw1 (C==9 only)
  __shared__ float wk[COLS];                      // softmax attention weights
  __shared__ float lg[COLS];                      // masked attention logits
  __shared__ float osum[64 * PTS];                // k-reduced output accum
  __shared__ int   nb[COLS];                      // neighbor indices

  const int tid  = threadIdx.x;
  const int lane = tid & 31;
  const int b    = blockIdx.y;
  const int n0   = blockIdx.x * PTS;
  const float* fb = fin + (long)b * finBS;

  if (tid < COLS) nb[tid] = idx[((long)b * N + (n0 + tid / KNN)) * KNN + (tid % KNN)];
  osum[tid] = 0.f;  // exactly 256 entries
  if constexpr (PADDED) {                         // stage zero-padded weights
    for (int t = tid; t < 64 * KP; t += 256) {
      int r = t / KP, c = t % KP;
      W1p[t] = (c < TWOC) ? tw1[r * TWOC + c] : 0.f;
    }
  }
  __syncthreads();

  // Stage edge features: rows [0,C)=neigh-center, [C,2C)=center, pad rows = 0
  for (int t = tid; t < KP * COLS; t += 256) {
    int kc = t / COLS, col = t % COLS;
    int p = col / KNN;
    float v = 0.f;
    if (kc < C)         v = fb[(long)kc * N + nb[col]] - fb[(long)kc * N + (n0 + p)];
    else if (kc < TWOC) v = fb[(long)(kc - C) * N + (n0 + p)];
    E[kc * COLS + col] = v;
  }
  __syncthreads();

  // Gate + attention branches (8 hidden channels; VALU — too skinny for WMMA)
  if (tid < COLS) {
    int p = tid / KNN, k = tid % KNN;
    float gl = gbias[0], al = abias[0];
    for (int o = 0; o < 8; ++o) {
      float sg = 0.f, sa = 0.f;
#pragma unroll 4
      for (int c = 0; c < TWOC; ++c) {
        float e = E[c * COLS + tid];
        sg += gw1[o * TWOC + c] * e;
        sa += aw1[o * TWOC + c] * e;
      }
      gl += gw2[o] * lrelu(gg1[o] * INVBN * sg + gb1[o]);
      al += aw2[o] * lrelu(ag1[o] * INVBN * sa + ab1[o]);
    }
    unsigned h = pcg_hash(((unsigned)((b * N + n0 + p) * KNN + k)) ^ seed);
    float ys = 1.f / (1.f + expf(-(gl + gumbel_noise(h))));
    lg[tid] = (ys >= 0.5f) ? al : NEGL;
  }
  __syncthreads();
  if (tid < PTS) {  // softmax over k=20 per point
    float mx = -3.0e38f;
    for (int k = 0; k < KNN; ++k) mx = fmaxf(mx, lg[tid * KNN + k]);
    float s = 0.f;
    for (int k = 0; k < KNN; ++k) { float e = expf(lg[tid * KNN + k] - mx); wk[tid * KNN + k] = e; s += e; }
    float r = 1.f / s;
    for (int k = 0; k < KNN; ++k) wk[tid * KNN + k] *= r;
  }
  __syncthreads();

  const int l16 = lane & 15;
  const int hi  = lane >> 4;                      // 0 | 1 (upper half-wave)
  // SGPR wave id -> scalar branches (no EXEC masking around WMMA)
  const int waveS = __builtin_amdgcn_readfirstlane(tid >> 5);
  const int m0    = (waveS & 3) * 16;             // M-tile
  const int c0    = (waveS >> 2) * 16;            // column tiles c0, c0+32 [,64]
  const bool third = (waveS >> 2) == 0;           // waves 0-3 own the 5th tile

  // ---- t-branch conv1: [64 x 2C] * [2C x 80] -----------------------------
  {
    v8f a0 = {}, a1 = {}, a2 = {};
#pragma unroll
    for (int kk = 0; kk < KP; kk += 4) {
      const int kr = kk + hi * 2;
      v2f af;
      if constexpr (PADDED) {
        af.x = W1p[(m0 + l16) * KP + kr];
        af.y = W1p[(m0 + l16) * KP + kr + 1];
      } else {
        af.x = tw1[(m0 + l16) * TWOC + kr];
        af.y = tw1[(m0 + l16) * TWOC + kr + 1];
      }
      v2f b0; b0.x = E[kr * COLS + c0 + l16];       b0.y = E[(kr + 1) * COLS + c0 + l16];
      a0 = wmma4(af, b0, a0);
      v2f b1; b1.x = E[kr * COLS + c0 + 32 + l16];  b1.y = E[(kr + 1) * COLS + c0 + 32 + l16];
      a1 = wmma4(af, b1, a1);
      if (third) {
        v2f b2; b2.x = E[kr * COLS + 64 + l16];     b2.y = E[(kr + 1) * COLS + 64 + l16];
        a2 = wmma4(af, b2, a2);
      }
    }
    auto epi = [&](v8f& acc, int cbase) {
#pragma unroll
      for (int v = 0; v < 8; ++v) {
        int m = m0 + v + hi * 8;
        float val = lrelu(tg1[m] * INVBN * acc[v] + tb1[m]);
        int col = cbase + l16;
        if constexpr (TWO_TRANS) T[m * COLS + col] = val;
        else atomicAdd(&osum[m * PTS + col / KNN], val * wk[col]);
      }
    };
    epi(a0, c0); epi(a1, c0 + 32); if (third) epi(a2, 64);
  }

  // ---- t-branch conv2: [64 x 64] * [64 x 80] from LDS-resident T ---------
  if constexpr (TWO_TRANS) {
    __syncthreads();
    v8f a0 = {}, a1 = {}, a2 = {};
#pragma unroll
    for (int kk = 0; kk < 64; kk += 4) {
      const int kr = kk + hi * 2;
      v2f af;
      af.x = tw2[(m0 + l16) * 64 + kr];
      af.y = tw2[(m0 + l16) * 64 + kr + 1];
      v2f b0; b0.x = T[kr * COLS + c0 + l16];       b0.y = T[(kr + 1) * COLS + c0 + l16];
      a0 = wmma4(af, b0, a0);
      v2f b1; b1.x = T[kr * COLS + c0 + 32 + l16];  b1.y = T[(kr + 1) * COLS + c0 + 32 + l16];
      a1 = wmma4(af, b1, a1);
      if (third) {
        v2f b2; b2.x = T[kr * COLS + 64 + l16];     b2.y = T[(kr + 1) * COLS + 64 + l16];
        a2 = wmma4(af, b2, a2);
      }
    }
    auto epi2 = [&](v8f& acc, int cbase) {
#pragma unroll
      for (int v = 0; v < 8; ++v) {
        int m = m0 + v + hi * 8;
        float val = lrelu(tg2[m] * INVBN * acc[v] + tb2[m]);
        int col = cbase + l16;
        atomicAdd(&osum[m * PTS + col / KNN], val * wk[col]);
      }
    };
    epi2(a0, c0); epi2(a1, c0 + 32); if (third) epi2(a2, 64);
  }
  __syncthreads();
  { // 64*PTS == 256: one output element per thread
    int m = tid / PTS, p = tid % PTS;
    out[(long)b * outBS + (long)m * N + (n0 + p)] = osum[tid];
  }
}

// ---------------------------------------------------------------------------
// Generic 1x1-conv GEMM: Y[b,m,n] = act(bn(W[m,:]*X[b,:,n] (+ base[b,m])))
// 8 waves/WG share one 16-row W panel staged in LDS; each wave one 16x16 tile.
// ---------------------------------------------------------------------------
template <int KD>
__global__ void __launch_bounds__(256)
gemm_bn_act_kernel(const float* __restrict__ W, int lda,
                   const float* __restrict__ X, long xBS,
                   float* __restrict__ Y, long yBS,
                   const float* __restrict__ gamma, const float* __restrict__ beta,
                   const float* __restrict__ baseVec, int baseLd,
                   int M, int N, int doLrelu) {
  __shared__ float Wl[16 * KD];
  const int tid  = threadIdx.x;
  const int lane = tid & 31;
  const int wave = tid >> 5;
  const int l16  = lane & 15;
  const int hi   = lane >> 4;
  const int z    = blockIdx.z;
  const int m0   = blockIdx.y * 16;
  const int n    = blockIdx.x * 128 + wave * 16 + l16;
  const float* Xb = X + (long)z * xBS;

  for (int t = tid; t < 16 * KD / 4; t += 256) {  // stage weight panel (b128)
    int r = (t * 4) / KD, c = (t * 4) % KD;       // KD % 4 == 0: no row cross
    int row = m0 + r; if (row >= M) row = M - 1;
    *(float4*)&Wl[r * KD + c] = *(const float4*)&W[(long)row * lda + c];
  }
  __syncthreads();

  v8f acc = {};
#pragma unroll 8
  for (int kk = 0; kk < KD; kk += 4) {
    const int kr = kk + hi * 2;
    v2f a, bf;
    a.x  = Wl[l16 * KD + kr];
    a.y  = Wl[l16 * KD + kr + 1];
    bf.x = Xb[(long)kr * N + n];
    bf.y = Xb[(long)(kr + 1) * N + n];
    if (kk + 16 < KD) __builtin_prefetch(&Xb[(long)(kr + 16) * N + n], 0, 0);
    acc = wmma4(a, bf, acc);
  }
#pragma unroll
  for (int v = 0; v < 8; ++v) {
    int m = m0 + v + hi * 8;
    if (m >= M) continue;
    float val = acc[v];
    if (baseVec) val += baseVec[(long)z * baseLd + m];
    if (gamma)     val = gamma[m] * INVBN * val + beta[m];
    else if (beta) val += beta[m];
    if (doLrelu) val = lrelu(val);
    Y[(long)z * yBS + (long)m * N + n] = val;
  }
}

// Global max over N per (b,channel) row (b128 loads)
__global__ void __launch_bounds__(256)
rowmax_kernel(const float* __restrict__ X, float* __restrict__ out, int N) {
  __shared__ float r[256];
  const long row = blockIdx.x;
  float m = -3.0e38f;
  for (int j = threadIdx.x * 4; j < N; j += 1024) {
    float4 v = *(const float4*)&X[row * N + j];
    m = fmaxf(m, fmaxf(fmaxf(v.x, v.y), fmaxf(v.z, v.w)));
  }
  r[threadIdx.x] = m;
  __syncthreads();
  for (int s = 128; s > 0; s >>= 1) {
    if (threadIdx.x < s) r[threadIdx.x] = fmaxf(r[threadIdx.x], r[threadIdx.x + s]);
    __syncthreads();
  }
  if (threadIdx.x == 0) out[row] = r[0];
}

// base[b,m] = sum_c<Kd W[m,c]*fu[b,c]  (fu part of pw1, n-independent)
__global__ void __launch_bounds__(256)
fu_base_kernel(const float* __restrict__ W, int lda,
               const float* __restrict__ fu, float* __restrict__ base,
               int M, int Kd) {
  __shared__ float fs[1024];
  const int b = blockIdx.x;
  for (int i = threadIdx.x; i < Kd; i += 256) fs[i] = fu[(long)b * Kd + i];
  __syncthreads();
  for (int m = threadIdx.x; m < M; m += 256) {
    float s = 0.f;
    for (int c = 0; c < Kd; c += 4) {
      float4 w4 = *(const float4*)&W[(long)m * lda + c];
      s += w4.x * fs[c] + w4.y * fs[c + 1] + w4.z * fs[c + 2] + w4.w * fs[c + 3];
    }
    base[(long)b * M + m] = s;
  }
}

// ---------------------------------------------------------------------------
extern "C" void kernel_launch(void* const* d_in, const int* in_sizes, int n_in,
                              void* d_out, int out_size, void* d_ws, size_t ws_size,
                              hipStream_t stream) {
  (void)in_sizes; (void)n_in; (void)out_size; (void)ws_size;
  const int N = 4096, Bn = 4;
  auto in = [&](int i) { return (const float*)d_in[i]; };

  const float* x = in(0);
  // block param order: tw1,tg1,tb1,gw1,gg1,gb1,gw2,gbias,aw1,ag1,ab1,aw2,abias[,tw2,tg2,tb2]
  const int B1 = 1, B2 = 17, B3 = 33;
  const float* fw  = in(46); const float* fg  = in(47); const float* fbv = in(48);
  const float* pw1 = in(49); const float* pg1 = in(50); const float* pb1 = in(51);
  const float* pw2 = in(52); const float* pg2 = in(53); const float* pb2 = in(54);
  const float* pw3 = in(55); const float* pbias = in(56);

  // workspace layout (floats)
  float* ws = (float*)d_ws;
  size_t off = 0;
  float* feats = ws + off; off += (size_t)Bn * 192 * N;   // [B,192,N]
  float* fu    = ws + off; off += (size_t)Bn * 1024;
  float* basev = ws + off; off += (size_t)Bn * 512;
  int*   idx1  = (int*)(ws + off); off += (size_t)Bn * N * KNN;
  int*   idx2  = (int*)(ws + off); off += (size_t)Bn * N * KNN;
  int*   idx3  = (int*)(ws + off); off += (size_t)Bn * N * KNN;
  float* big   = ws + off;                                 // fuout aliases h1/h2
  float* fuout = big;                                      // [B,1024,N]
  float* h1    = big;                                      // [B,512,N] (after fu done)
  float* h2    = big + (size_t)Bn * 512 * N;               // [B,256,N]

  dim3 blk(256);

  // 1) kNN on xyz -> idx1 ; edge block 1 -> feats[0:64]
  knn_kernel<3><<<dim3(Bn * N), blk, 0, stream>>>(x, (long)9 * N, N, idx1);
  edge_block_kernel<9, true><<<dim3(N / PTS, Bn), blk, 0, stream>>>(
      x, (long)9 * N, idx1,
      in(B1+0), in(B1+1), in(B1+2), in(B1+13), in(B1+14), in(B1+15),
      in(B1+3), in(B1+4), in(B1+5), in(B1+6), in(B1+7),
      in(B1+8), in(B1+9), in(B1+10), in(B1+11), in(B1+12),
      feats, (long)192 * N, N, 0xA341u);

  // 2) kNN on n1 ; edge block 2 -> feats[64:128]
  knn_kernel<64><<<dim3(Bn * N), blk, 0, stream>>>(feats, (long)192 * N, N, idx2);
  edge_block_kernel<64, true><<<dim3(N / PTS, Bn), blk, 0, stream>>>(
      feats, (long)192 * N, idx2,
      in(B2+0), in(B2+1), in(B2+2), in(B2+13), in(B2+14), in(B2+15),
      in(B2+3), in(B2+4), in(B2+5), in(B2+6), in(B2+7),
      in(B2+8), in(B2+9), in(B2+10), in(B2+11), in(B2+12),
      feats + (size_t)64 * N, (long)192 * N, N, 0xB713u);

  // 3) kNN on n2 ; edge block 3 (single trans conv) -> feats[128:192]
  knn_kernel<64><<<dim3(Bn * N), blk, 0, stream>>>(feats + (size_t)64 * N, (long)192 * N, N, idx3);
  edge_block_kernel<64, false><<<dim3(N / PTS, Bn), blk, 0, stream>>>(
      feats + (size_t)64 * N, (long)192 * N, idx3,
      in(B3+0), in(B3+1), in(B3+2), nullptr, nullptr, nullptr,
      in(B3+3), in(B3+4), in(B3+5), in(B3+6), in(B3+7),
      in(B3+8), in(B3+9), in(B3+10), in(B3+11), in(B3+12),
      feats + (size_t)128 * N, (long)192 * N, N, 0xC957u);

  // 4) fw conv [1024x192] + bn + lrelu -> fuout ; global max -> fu
  gemm_bn_act_kernel<192><<<dim3(N / 128, 1024 / 16, Bn), blk, 0, stream>>>(
      fw, 192, feats, (long)192 * N, fuout, (long)1024 * N,
      fg, fbv, nullptr, 0, 1024, N, 1);
  rowmax_kernel<<<dim3(Bn * 1024), blk, 0, stream>>>(fuout, fu, N);

  // 5) pw1 split: fu part as per-batch base vector, feats part as GEMM
  fu_base_kernel<<<dim3(Bn), blk, 0, stream>>>(pw1, 1216, fu, basev, 512, 1024);
  gemm_bn_act_kernel<192><<<dim3(N / 128, 512 / 16, Bn), blk, 0, stream>>>(
      pw1 + 1024, 1216, feats, (long)192 * N, h1, (long)512 * N,
      pg1, pb1, basev, 512, 512, N, 1);

  // 6) pw2 [256x512] + bn + lrelu
  gemm_bn_act_kernel<512><<<dim3(N / 128, 256 / 16, Bn), blk, 0, stream>>>(
      pw2, 512, h1, (long)512 * N, h2, (long)256 * N,
      pg2, pb2, nullptr, 0, 256, N, 1);

  // 7) pw3 [13x256] + bias -> d_out [B,13,N]
  gemm_bn_act_kernel<256><<<dim3(N / 128, 1, Bn), blk, 0, stream>>>(
      pw3, 256, h2, (long)256 * N, (float*)d_out, (long)13 * N,
      nullptr, pbias, nullptr, 0, 13, N, 0);
}